// Shift3x3_6794638262361
// MI455X (gfx1250) — compile-verified
//
#include <hip/hip_runtime.h>
#include <hip/hip_bf16.h>
#include <stdint.h>

// Shift3x3 channel-shift kernel for MI455X (gfx1250).
// Memory-bound permutation: stage each (b,c) 56x56 fp32 plane into LDS via the
// CDNA5 async global->LDS path (ASYNCcnt), then emit shifted, aligned b128
// non-temporal stores. wave32, 256 threads (8 waves) per block.

#define B_ 32
#define C_ 384
#define H_ 56
#define W_ 56
#define PLANE (H_ * W_)        // 3136 floats
#define PLANE_V4 (PLANE / 4)   // 784 float4s

typedef float v4f __attribute__((ext_vector_type(4)));
typedef int   v4i __attribute__((ext_vector_type(4)));

#define GLOBAL_AS __attribute__((address_space(1)))
#define LDS_AS    __attribute__((address_space(3)))

#if __has_builtin(__builtin_amdgcn_global_load_async_to_lds_b128) && \
    __has_builtin(__builtin_amdgcn_s_wait_asynccnt)
#define USE_ASYNC_LDS 1
#else
#define USE_ASYNC_LDS 0
#endif

__global__ __launch_bounds__(256) void shift3x3_kernel(
    const float* __restrict__ x, float* __restrict__ out) {
  __shared__ float tile[PLANE];  // 12544 bytes of the 320KB/WGP LDS

  const int plane = blockIdx.x;          // plane = b * C + c
  const int c = plane % C_;
  const int type = c % 5;                // uniform across the block
  const size_t base = (size_t)plane * PLANE;
  const float* __restrict__ src = x + base;
  float* __restrict__ dst = out + base;
  const int tid = threadIdx.x;

  // ---- Stage this plane into LDS -------------------------------------------
#if USE_ASYNC_LDS
  // Each lane moves 16B memory->LDS directly (no VGPR round-trip); one
  // global_load_async_to_lds_b128 per wave-iteration, tracked by ASYNCcnt.
  for (int q = tid; q < PLANE_V4; q += 256) {
    __builtin_amdgcn_global_load_async_to_lds_b128(
        (GLOBAL_AS v4i*)(src + (size_t)q * 4),
        (LDS_AS v4i*)&tile[q * 4],
        /*imm offset*/ 0, /*cpol*/ 0);
  }
  __builtin_amdgcn_s_wait_asynccnt(0);
#else
  for (int q = tid; q < PLANE_V4; q += 256) {
    v4f v = __builtin_nontemporal_load((const v4f*)(src + (size_t)q * 4));
    *(v4f*)&tile[q * 4] = v;
  }
#endif
  __syncthreads();

  // ---- Emit shifted output: one aligned float4 per (thread, iteration) -----
  for (int q = tid; q < PLANE_V4; q += 256) {
    const int i = q / (W_ / 4);        // row 0..55
    const int j = (q % (W_ / 4)) * 4;  // col 0,4,...,52
    v4f v;
    if (type == 0) {                   // out[i][j] = x[i-1][j]
      v = (i > 0) ? *(const v4f*)&tile[(i - 1) * W_ + j] : (v4f)(0.0f);
    } else if (type == 1) {            // out[i][j] = x[i][j-1]
      const int r = i * W_;
      v.x = (j > 0) ? tile[r + j - 1] : 0.0f;
      v.y = tile[r + j + 0];
      v.z = tile[r + j + 1];
      v.w = tile[r + j + 2];
    } else if (type == 2) {            // identity
      v = *(const v4f*)&tile[i * W_ + j];
    } else if (type == 3) {            // out[i][j] = x[i][j+1]
      const int r = i * W_;
      v.x = tile[r + j + 1];
      v.y = tile[r + j + 2];
      v.z = tile[r + j + 3];
      v.w = (j + 4 < W_) ? tile[r + j + 4] : 0.0f;
    } else {                           // type == 4: out[i][j] = x[i+1][j]
      v = (i < H_ - 1) ? *(const v4f*)&tile[(i + 1) * W_ + j] : (v4f)(0.0f);
    }
    // Streaming store: read-once/write-once data, keep it out of L2 residency.
    __builtin_nontemporal_store(v, (v4f*)(dst + i * W_ + j));
  }
}

extern "C" void kernel_launch(void* const* d_in, const int* in_sizes, int n_in,
                              void* d_out, int out_size, void* d_ws, size_t ws_size,
                              hipStream_t stream) {
  (void)in_sizes; (void)n_in; (void)out_size; (void)d_ws; (void)ws_size;
  const float* x = (const float*)d_in[0];
  float* out = (float*)d_out;
  shift3x3_kernel<<<dim3(B_ * C_), dim3(256), 0, stream>>>(x, out);
}